// GraphConvolution_66778151518716
// MI455X (gfx1250) — compile-verified
//
#include <hip/hip_runtime.h>

#define BDIM   2
#define NNODES 50000
#define NEDGES 800000
#define DIN    128
#define DOUT   128
#define NCOLS  (BDIM * DOUT)   // 256 node-major columns

typedef __attribute__((ext_vector_type(2))) float v2f;
typedef __attribute__((ext_vector_type(8))) float v8f;

// ---------------------------------------------------------------------------
// Kernel 1: support_nm[n, b*128+o] = sum_k x[b,n,k] * w[k,o]
// One wave32 per 16x16 output tile, fp32 WMMA, K=128 in 32 steps of 4.
// Tiles: (N/16)=3125 node tiles x 16 column tiles = 50000 waves exactly.
// ---------------------------------------------------------------------------
__global__ void gcn_gemm_wmma(const float* __restrict__ x,
                              const float* __restrict__ w,
                              float* __restrict__ sup) {
  const int wavesPerBlk = blockDim.x >> 5;
  const int tile = blockIdx.x * wavesPerBlk + (threadIdx.x >> 5);
  if (tile >= (NNODES / 16) * 16) return;   // wave-uniform guard (never taken)

  const int lane = threadIdx.x & 31;
  const int ct    = tile & 15;      // column tile 0..15
  const int nt    = tile >> 4;      // node tile 0..3124
  const int b     = ct >> 3;        // batch: columns [0,128) -> b=0, [128,256) -> b=1
  const int col0  = (ct & 7) * 16;  // column offset within DOUT
  const int node0 = nt * 16;

  const int hi  = lane >> 4;        // half-wave select
  const int l16 = lane & 15;

  // A: 16x4 fp32 tile. lane<16 holds (M=l16, K=4t+{0,1}), lane>=16 holds K=4t+{2,3}.
  const float* xrow = x + ((size_t)b * NNODES + node0 + l16) * DIN + 2 * hi;
  // B: 4x16 fp32 tile. lane<16 holds (K=4t+{0,1}, N=l16), lane>=16 K=4t+{2,3}.
  const float* wcol = w + (size_t)(2 * hi) * DOUT + (col0 + l16);

  v8f acc = {};
#pragma unroll
  for (int t = 0; t < 32; ++t) {
    v2f a = *(const v2f*)(xrow + 4 * t);          // contiguous b64 load
    v2f bb;
    bb.x = wcol[(size_t)(4 * t) * DOUT];          // K = 4t + 2*hi
    bb.y = wcol[(size_t)(4 * t + 1) * DOUT];      // K = 4t + 2*hi + 1
    acc = __builtin_amdgcn_wmma_f32_16x16x4_f32(
        /*neg_a=*/false, a, /*neg_b=*/false, bb,
        /*c_mod=*/(short)0, acc, /*reuse_a=*/false, /*reuse_b=*/false);
  }

  // D layout: VGPR r -> (M = r + 8*hi, N = l16)
  float* out = sup + ((size_t)(node0 + 8 * hi)) * NCOLS + b * DOUT + col0 + l16;
#pragma unroll
  for (int r = 0; r < 8; ++r)
    out[(size_t)r * NCOLS] = acc[r];
}

// ---------------------------------------------------------------------------
// Kernel 2: out[b,n,o] = bias[o]  (d_out is poisoned by the harness)
// float4-coalesced: 3.2M float4 = 12500 blocks x 256 threads exactly.
// ---------------------------------------------------------------------------
__global__ void gcn_init_bias(const float* __restrict__ bias,
                              float4* __restrict__ out) {
  size_t i = (size_t)blockIdx.x * blockDim.x + threadIdx.x;
  int o4 = (int)(i & (DOUT / 4 - 1));   // 32 float4 per 128-wide row
  out[i] = ((const float4*)bias)[o4];
}

// ---------------------------------------------------------------------------
// Kernel 3: edge scatter with L2 float atomics.
// One wave per edge: lane l handles 8 contiguous columns -> fully coalesced
// 1KB row read from support_nm, 8 atomic f32 adds into out[b, row, :].
// ---------------------------------------------------------------------------
__global__ void gcn_spmm(const int* __restrict__ rows,
                         const int* __restrict__ cols,
                         const float* __restrict__ vals,
                         const float* __restrict__ sup,
                         float* __restrict__ out) {
  const int e = blockIdx.x * (blockDim.x >> 5) + (threadIdx.x >> 5);
  if (e >= NEDGES) return;   // wave-uniform (grid is exact)

  const int lane = threadIdx.x & 31;
  const int r = rows[e];
  const int c = cols[e];
  const float v = vals[e];

  const float4* src = (const float4*)(sup + (size_t)c * NCOLS) + lane * 2;
  float4 s0 = src[0];
  float4 s1 = src[1];

  const int b  = lane >> 4;            // 8 cols per lane -> lane half picks batch
  const int o0 = (lane & 15) * 8;
  float* dst = out + ((size_t)b * NNODES + r) * DOUT + o0;

  atomicAdd(dst + 0, v * s0.x);
  atomicAdd(dst + 1, v * s0.y);
  atomicAdd(dst + 2, v * s0.z);
  atomicAdd(dst + 3, v * s0.w);
  atomicAdd(dst + 4, v * s1.x);
  atomicAdd(dst + 5, v * s1.y);
  atomicAdd(dst + 6, v * s1.z);
  atomicAdd(dst + 7, v * s1.w);
}

// ---------------------------------------------------------------------------
extern "C" void kernel_launch(void* const* d_in, const int* in_sizes, int n_in,
                              void* d_out, int out_size, void* d_ws, size_t ws_size,
                              hipStream_t stream) {
  const float* x    = (const float*)d_in[0];  // [B, N, DIN]
  const int*   rows = (const int*)d_in[1];    // [E]
  const int*   cols = (const int*)d_in[2];    // [E]
  const float* vals = (const float*)d_in[3];  // [E]
  const float* w    = (const float*)d_in[4];  // [DIN, DOUT]
  const float* bias = (const float*)d_in[5];  // [DOUT]
  float* out = (float*)d_out;                 // [B, N, DOUT]
  float* sup = (float*)d_ws;                  // support_nm [N, 256] = 51.2 MB

  // 1) fp32 WMMA GEMM -> node-major support matrix in scratch
  //    50000 waves, 4 waves per 128-thread block -> 12500 blocks
  hipLaunchKernelGGL(gcn_gemm_wmma, dim3(12500), dim3(128), 0, stream, x, w, sup);

  // 2) broadcast bias into output (3.2M float4)
  hipLaunchKernelGGL(gcn_init_bias, dim3(12500), dim3(256), 0, stream, bias,
                     (float4*)out);

  // 3) atomic edge scatter: 8 waves (edges) per 256-thread block
  hipLaunchKernelGGL(gcn_spmm, dim3(NEDGES / 8), dim3(256), 0, stream, rows,
                     cols, vals, sup, out);
}